// EMP_Net_75728863363104
// MI455X (gfx1250) — compile-verified
//
#include <hip/hip_runtime.h>

// OTAM cumulative soft-DTW distance for MI455X (gfx1250, wave32).
// One lane per (q,s) pair; 34-entry DP row held in VGPRs (unrolled column
// loop). Each wave stages its 32 pair-rows (4KB/row-step) into LDS with
// coalesced async global->LDS b128 copies, double buffered one row ahead
// (ASYNCcnt), padded to 36 floats/row to avoid LDS bank conflicts.

#define QDIM   200
#define SDIM   200
#define LROWS  32
#define MCOLS  32
#define PAIRS  (QDIM * SDIM)      // 40000
#define WAVES  2                  // waves per block
#define TPB    (WAVES * 32)       // 64 threads
#define NBLK   (PAIRS / TPB)      // 625 blocks, exact
#define ROWPAD 36                 // floats per pair-row in LDS (144B, 16B aligned)

static_assert(PAIRS % TPB == 0, "exact tiling");

#define A_LN2 0.34657359027997264f   // lambda * ln(2), lambda = 0.5
#define INV_A 2.8853900817779268f    // 1 / (lambda * ln 2)

#if defined(__AMDGCN__) && __has_builtin(__builtin_amdgcn_global_load_async_to_lds_b128)
#define OTAM_ASYNC 1
#else
#define OTAM_ASYNC 0
#endif

typedef int v4i __attribute__((ext_vector_type(4)));
typedef v4i __attribute__((address_space(1))) global_v4i;
typedef v4i __attribute__((address_space(3))) lds_v4i;

__device__ __forceinline__ void copy16_issue(const float* __restrict__ g, float* l) {
#if OTAM_ASYNC
  // gfx1250 async DMA: global -> LDS, 16 bytes per lane, tracked by ASYNCcnt.
  __builtin_amdgcn_global_load_async_to_lds_b128(
      (global_v4i*)g,
      (lds_v4i*)l,
      /*offset=*/0, /*cpol=*/0);
#else
  *(float4*)l = *(const float4*)g;   // synchronous fallback, same coalescing
#endif
}

__device__ __forceinline__ void copy_wait() {
#if OTAM_ASYNC
#if __has_builtin(__builtin_amdgcn_s_wait_asynccnt)
  __builtin_amdgcn_s_wait_asynccnt(0);
#else
  asm volatile("s_wait_asynccnt 0x0" ::: "memory");
#endif
  asm volatile("" ::: "memory");     // keep LDS reads below the wait
#endif
}

// Stable base-2 softmin, lambda = 0.5:
//   softmin(x..) = -lambda * ln(sum exp(-x/lambda))
//                =  min    - A_LN2 * log2(sum 2^(-(x-min)*INV_A))
__device__ __forceinline__ float softmin2(float x, float y) {
  float mn = fminf(x, y);
  float dd = fabsf(x - y);
  return mn - A_LN2 * log2f(1.0f + exp2f(-dd * INV_A));
}
__device__ __forceinline__ float softmin3(float x, float y, float z) {
  float mn = fminf(fminf(x, y), z);
  float s  = exp2f((mn - x) * INV_A)
           + exp2f((mn - y) * INV_A)
           + exp2f((mn - z) * INV_A);
  return mn - A_LN2 * log2f(s);
}

__global__ __launch_bounds__(TPB) void otam_kernel(const float* __restrict__ dists,
                                                   float* __restrict__ out) {
  __shared__ float smem[2][WAVES][32 * ROWPAD];  // double-buffered row stage

  const int lane     = threadIdx.x & 31;
  const int wv       = threadIdx.x >> 5;
  const int waveBase = (blockIdx.x * WAVES + wv) * 32;       // first pair of this wave
  const float* gwave = dists + (size_t)waveBase * (LROWS * MCOLS);

  // Coalesced copy of one matrix row (32 floats) for each of the wave's 32
  // pairs: instruction k moves pairs k*4..k*4+3; lanes 0-7 cover one pair's
  // 128B row contiguously (16B per lane).
  const int pk0 = lane >> 3;        // 0..3
  const int ch4 = (lane & 7) * 4;   // float offset of this lane's 16B chunk

  auto issue = [&](int buf, int l) {
    float* lbuf = &smem[buf][wv][0];
#pragma unroll
    for (int k = 0; k < 8; ++k) {
      int pk = k * 4 + pk0;
      copy16_issue(gwave + (size_t)pk * (LROWS * MCOLS) + l * MCOLS + ch4,
                   lbuf + pk * ROWPAD + ch4);
    }
  };

  issue(0, 0);
  copy_wait();       // row 0 resident
  issue(1, 1);       // prefetch row 1 while we do the cumsum

  // prev[m], m = 0..33: padded DP row, kept in VGPRs (constant indices only).
  float prev[34];
  prev[0] = 0.0f;    // padded column 0 is never written
  {
    const float* drow0 = &smem[0][wv][lane * ROWPAD];
    float acc = 0.0f;
#pragma unroll
    for (int m = 1; m <= 32; ++m) { acc += drow0[m - 1]; prev[m] = acc; }
    prev[33] = acc;  // + pad(0)
  }

  for (int l = 1; l < LROWS; ++l) {
    copy_wait();                              // row l resident in smem[l&1]
    if (l + 1 < LROWS) issue((l + 1) & 1, l + 1);

    const float* drow = &smem[l & 1][wv][lane * ROWPAD];
    float c    = 0.0f;        // running cum[l][m-1]; starts at padded col 0 = 0
    float diag = prev[0];     // old prev[m-1]
#pragma unroll
    for (int m = 1; m <= 33; ++m) {
      float up = prev[m];     // old prev[m] (before overwrite)
      float sm = (m == 1 || m == 33) ? softmin3(diag, c, up)
                                     : softmin2(diag, c);
      float dm = (m <= 32) ? drow[m - 1] : 0.0f;   // padded col 33 = 0
      float v  = dm + sm;
      prev[m]  = v;
      c        = v;
      diag     = up;
    }
  }

  out[waveBase + lane] = prev[33];   // cum_dists[.., -1, -1]
}

extern "C" void kernel_launch(void* const* d_in, const int* in_sizes, int n_in,
                              void* d_out, int out_size, void* d_ws, size_t ws_size,
                              hipStream_t stream) {
  (void)in_sizes; (void)n_in; (void)d_ws; (void)ws_size; (void)out_size;
  const float* dists = (const float*)d_in[0];
  float* out = (float*)d_out;
  otam_kernel<<<NBLK, TPB, 0, stream>>>(dists, out);
}